// GATsimple_8821862826463
// MI455X (gfx1250) — compile-verified
//
#include <hip/hip_runtime.h>
#include <cstdint>
#include <cstddef>

typedef float v2f __attribute__((ext_vector_type(2)));
typedef float v8f __attribute__((ext_vector_type(8)));

#define N_NODES 17024
#define N_EDGES 136192
#define E_TOT   (N_EDGES + N_NODES)   // reference appends one self-loop per node
#define HEADS   4

// ---------------------------------------------------------------- utilities

__global__ void fill_f32(float* __restrict__ p, float v, int n) {
  int i = blockIdx.x * blockDim.x + threadIdx.x;
  if (i < n) p[i] = v;
}

__device__ __forceinline__ void atomicMaxF(float* addr, float val) {
  // sign-split trick: float ordering == int ordering for >=0,
  // reversed unsigned ordering for <0. Mixed-sign races resolve correctly.
  if (val >= 0.0f) atomicMax((int*)addr, __float_as_int(val));
  else             atomicMin((unsigned int*)addr, __float_as_uint(val));
}

// ------------------------------------------------- fp32 WMMA GEMM: H = X @ W
// X: [N, Fin] row-major, W: [Fin, Fout] row-major, Hout: [N, Fout].
// One wave per 16x16 output tile, K-loop in steps of 4 via V_WMMA_F32_16X16X4_F32.
// Fragment layouts (CDNA5 ISA 7.12.2):
//   A 16x4 : lanes0-15 = rows, vgpr{0,1} = K{0,1}; lanes16-31 = K{2,3}
//   B 4x16 : lanes0-15 = cols for K{0,1}; lanes16-31 = cols for K{2,3}
//   C 16x16: vgpr r -> row r (lanes0-15) / row r+8 (lanes16-31), col = lane&15
__global__ void gemm_wmma_f32(const float* __restrict__ X, const float* __restrict__ W,
                              float* __restrict__ Hout, int Fin, int Fout,
                              int tilesM, int tilesN) {
  int wave = (int)((blockIdx.x * blockDim.x + threadIdx.x) >> 5);
  if (wave >= tilesM * tilesN) return;           // wave-uniform: scalar branch, EXEC stays all-1
  int lane = (int)(threadIdx.x & 31);
  int tm = wave % tilesM;
  int tn = wave / tilesM;
  int hi  = lane >> 4;                            // 0: lanes 0-15, 1: lanes 16-31
  int l15 = lane & 15;
  int arow = tm * 16 + l15;                       // A row this lane supplies
  int bcol = tn * 16 + l15;                       // B/C column this lane supplies
  int kb   = hi * 2;                              // K sub-offset (0 or 2)

  v8f acc = {};
  for (int k0 = 0; k0 < Fin; k0 += 4) {
    v2f a = *(const v2f*)(X + (size_t)arow * Fin + (k0 + kb));      // K{kb,kb+1}
    v2f b;
    b.x = W[(size_t)(k0 + kb    ) * Fout + bcol];
    b.y = W[(size_t)(k0 + kb + 1) * Fout + bcol];
    acc = __builtin_amdgcn_wmma_f32_16x16x4_f32(
        /*neg_a=*/false, a, /*neg_b=*/false, b,
        /*c_mod=*/(short)0, acc, /*reuse_a=*/false, /*reuse_b=*/false);
  }

  int m0 = tm * 16 + hi * 8;
#pragma unroll
  for (int r = 0; r < 8; ++r)
    Hout[(size_t)(m0 + r) * Fout + bcol] = acc[r];
}

// ------------------------------------------- per-node attention score dots
// s[n,h] = sum_c H[n, h*C+c] * a_src[h,c];  d likewise with a_dst.
__global__ void node_scores(const float* __restrict__ Hf,
                            const float* __restrict__ a_src,
                            const float* __restrict__ a_dst,
                            float* __restrict__ s, float* __restrict__ d, int C) {
  int tid = blockIdx.x * blockDim.x + threadIdx.x;
  if (tid >= N_NODES * HEADS) return;
  int n = tid / HEADS;
  int h = tid - n * HEADS;
  const float* hp = Hf + (size_t)n * (HEADS * C) + (size_t)h * C;
  const float* asv = a_src + (size_t)h * C;
  const float* adv = a_dst + (size_t)h * C;
  float accs = 0.0f, accd = 0.0f;
  for (int c = 0; c < C; ++c) {
    float v = hp[c];
    accs += v * asv[c];
    accd += v * adv[c];
  }
  s[tid] = accs;
  d[tid] = accd;
}

// ------------------------------------ edge pass 1: e = leakyrelu(s[src]+d[dst])
// store raw e; segment-max over dst via atomic float max.
__global__ void edge_scores(const int* __restrict__ ei,
                            const float* __restrict__ s, const float* __restrict__ d,
                            float* __restrict__ ebuf, float* __restrict__ emax) {
  int e = blockIdx.x * blockDim.x + threadIdx.x;
  if (e >= E_TOT) return;
  int src = (e < N_EDGES) ? ei[e]           : (e - N_EDGES);
  int dst = (e < N_EDGES) ? ei[N_EDGES + e] : (e - N_EDGES);
  float4 sv = ((const float4*)s)[src];
  float4 dv = ((const float4*)d)[dst];
  float ev[4] = { sv.x + dv.x, sv.y + dv.y, sv.z + dv.z, sv.w + dv.w };
  float4 out;
  float* op = &out.x;
#pragma unroll
  for (int h = 0; h < HEADS; ++h) {
    float v = ev[h];
    v = (v > 0.0f) ? v : 0.2f * v;          // LeakyReLU(0.2)
    op[h] = v;
    atomicMaxF(&emax[dst * HEADS + h], v);
  }
  ((float4*)ebuf)[e] = out;
}

// --------------------- edge pass 2: ee = exp(e - emax[dst]); denom += ee
__global__ void edge_exp(const int* __restrict__ ei,
                         float* __restrict__ ebuf,
                         const float* __restrict__ emax,
                         float* __restrict__ denom) {
  int e = blockIdx.x * blockDim.x + threadIdx.x;
  if (e >= E_TOT) return;
  int dst = (e < N_EDGES) ? ei[N_EDGES + e] : (e - N_EDGES);
  float4 ev = ((const float4*)ebuf)[e];
  float4 mv = ((const float4*)emax)[dst];
  ev.x = expf(ev.x - mv.x);
  ev.y = expf(ev.y - mv.y);
  ev.z = expf(ev.z - mv.z);
  ev.w = expf(ev.w - mv.w);
  ((float4*)ebuf)[e] = ev;
  atomicAdd(&denom[dst * HEADS + 0], ev.x);
  atomicAdd(&denom[dst * HEADS + 1], ev.y);
  atomicAdd(&denom[dst * HEADS + 2], ev.z);
  atomicAdd(&denom[dst * HEADS + 3], ev.w);
}

// --------------- edge pass 3: Out[dst] += (ee/denom[dst]) * H[src] (float4)
__global__ void edge_aggr(const int* __restrict__ ei,
                          const float* __restrict__ ebuf,
                          const float* __restrict__ denom,
                          const float* __restrict__ Hf,
                          float* __restrict__ Out,
                          int Fout, int C, int chunks) {
  long tid = (long)blockIdx.x * blockDim.x + threadIdx.x;
  if (tid >= (long)E_TOT * chunks) return;
  int e  = (int)(tid / chunks);
  int ch = (int)(tid - (long)e * chunks);
  int c0 = ch * 4;
  int head = c0 / C;
  int src = (e < N_EDGES) ? ei[e]           : (e - N_EDGES);
  int dst = (e < N_EDGES) ? ei[N_EDGES + e] : (e - N_EDGES);
  float alpha = ebuf[e * HEADS + head] / denom[dst * HEADS + head];
  float4 hv = *(const float4*)(Hf + (size_t)src * Fout + c0);
  float* o = Out + (size_t)dst * Fout + c0;
  atomicAdd(o + 0, alpha * hv.x);
  atomicAdd(o + 1, alpha * hv.y);
  atomicAdd(o + 2, alpha * hv.z);
  atomicAdd(o + 3, alpha * hv.w);
}

// ---------------------------------------------- fused bias + ELU (alpha=1)
__global__ void bias_elu(const float* __restrict__ In, const float* __restrict__ bias,
                         float* __restrict__ OutX, int Fout, int total) {
  int tid = blockIdx.x * blockDim.x + threadIdx.x;
  if (tid >= total) return;
  int f = tid % Fout;
  float v = In[tid] + bias[f];
  OutX[tid] = (v > 0.0f) ? v : (expf(v) - 1.0f);
}

// ------------------------------- readout: [128, 8512] @ fcw[8512,1] + fcb
__global__ void fc_out(const float* __restrict__ X, const float* __restrict__ w,
                       const float* __restrict__ b, float* __restrict__ out) {
  __shared__ float red[256];
  int row = blockIdx.x;
  int t = threadIdx.x;
  const float* xr = X + (size_t)row * 8512;
  float acc = 0.0f;
  for (int i = t; i < 8512; i += 256) acc += xr[i] * w[i];
  red[t] = acc;
  __syncthreads();
  for (int s = 128; s > 0; s >>= 1) {
    if (t < s) red[t] += red[t + s];
    __syncthreads();
  }
  if (t == 0) out[row] = red[0] + b[0];
}

// --------------------------------------------------------------- launcher

extern "C" void kernel_launch(void* const* d_in, const int* in_sizes, int n_in,
                              void* d_out, int out_size, void* d_ws, size_t ws_size,
                              hipStream_t stream) {
  (void)in_sizes; (void)n_in; (void)out_size; (void)ws_size;

  const float* x  = (const float*)d_in[0];
  const int*   ei = (const int*)d_in[1];
  // d_in[2] = batch (layout is static; unused)
  const float *Wm[4], *As[4], *Ad[4], *Bs[4];
  for (int i = 0; i < 4; ++i) {
    Wm[i] = (const float*)d_in[3 + 4 * i];
    As[i] = (const float*)d_in[4 + 4 * i];
    Ad[i] = (const float*)d_in[5 + 4 * i];
    Bs[i] = (const float*)d_in[6 + 4 * i];
  }
  const float* fcw = (const float*)d_in[19];
  const float* fcb = (const float*)d_in[20];

  float* ws = (float*)d_ws;
  const size_t NF = (size_t)N_NODES * 512;          // max per-node feature width
  const size_t NH = (size_t)N_NODES * HEADS;
  float* bufA = ws;                                  // layer activations (ELU output)
  float* bufB = bufA + NF;                           // h = x @ W
  float* bufC = bufB + NF;                           // aggregation accumulator
  float* sS   = bufC + NF;                           // [N,H] src scores
  float* sD   = sS + NH;                             // [N,H] dst scores
  float* sMax = sD + NH;                             // [N,H] segment max
  float* sDen = sMax + NH;                           // [N,H] softmax denom
  float* eBuf = sDen + NH;                           // [E_TOT,H] edge scratch

  const int FinA[4]  = { 64, 512, 256, 128 };
  const int FoutA[4] = { 512, 256, 128, 64 };
  const float* xin = x;
  const float NEG_INF = -__builtin_inff();

  for (int L = 0; L < 4; ++L) {
    const int fin = FinA[L], fout = FoutA[L], C = fout / HEADS;
    const int tilesM = N_NODES / 16, tilesN = fout / 16;
    const int waves = tilesM * tilesN;

    gemm_wmma_f32<<<(waves + 3) / 4, 128, 0, stream>>>(xin, Wm[L], bufB, fin, fout,
                                                       tilesM, tilesN);

    const int nh = N_NODES * HEADS;
    node_scores<<<(nh + 255) / 256, 256, 0, stream>>>(bufB, As[L], Ad[L], sS, sD, C);

    fill_f32<<<(nh + 255) / 256, 256, 0, stream>>>(sMax, NEG_INF, nh);
    fill_f32<<<(nh + 255) / 256, 256, 0, stream>>>(sDen, 0.0f, nh);
    const int nt = N_NODES * fout;
    fill_f32<<<(nt + 255) / 256, 256, 0, stream>>>(bufC, 0.0f, nt);

    edge_scores<<<(E_TOT + 255) / 256, 256, 0, stream>>>(ei, sS, sD, eBuf, sMax);
    edge_exp<<<(E_TOT + 255) / 256, 256, 0, stream>>>(ei, eBuf, sMax, sDen);

    const int chunks = fout / 4;
    const long tot3 = (long)E_TOT * chunks;
    edge_aggr<<<(int)((tot3 + 255) / 256), 256, 0, stream>>>(ei, eBuf, sDen, bufB, bufC,
                                                             fout, C, chunks);

    bias_elu<<<(nt + 255) / 256, 256, 0, stream>>>(bufC, Bs[L], bufA, fout, nt);
    xin = bufA;
  }

  fc_out<<<128, 256, 0, stream>>>(bufA, fcw, fcb, (float*)d_out);
}